// GTN_31628139168307
// MI455X (gfx1250) — compile-verified
//
#include <hip/hip_runtime.h>
#include <hip/hip_bf16.h>

// Problem constants (from the reference)
#define U_NODES   100000
#define I_NODES   40000
#define N_NODES   140000           // U + I
#define EMB       64
#define NEDGES    1000000
#define BATCHSZ   8192
#define KLAYERS   3
#define GAMMA     0.2f             // 1/(1+LAMBDA2), LAMBDA2=4
#define COEF      2.5f             // BETA/GAMMA = 0.5/0.2
#define LAM       4.0f             // clip bound

typedef __attribute__((ext_vector_type(2))) float v2f;
typedef __attribute__((ext_vector_type(8))) float v8f;

static inline int cdiv_h(long long a, long long b) { return (int)((a + b - 1) / b); }

// ---- degree / normalization --------------------------------------------------
__global__ void init_deg(float* __restrict__ deg) {
    int i = blockIdx.x * blockDim.x + threadIdx.x;
    if (i < N_NODES) deg[i] = 1.0f;                 // self-loop
}

__global__ void accum_deg(const int* __restrict__ eu, const int* __restrict__ ei,
                          float* __restrict__ deg) {
    int e = blockIdx.x * blockDim.x + threadIdx.x;
    if (e >= NEDGES) return;
    atomicAdd(&deg[U_NODES + ei[e]], 1.0f);
    atomicAdd(&deg[eu[e]], 1.0f);
}

__global__ void finish_dinv(float* __restrict__ deg) {
    int i = blockIdx.x * blockDim.x + threadIdx.x;
    if (i < N_NODES) deg[i] = rsqrtf(deg[i]);       // deg^-1/2, in place
}

// ---- node-wise kernels (float4 vectorized; N*EMB and U*EMB are /4) ----------
__global__ void init_x4(const float4* __restrict__ ue, const float4* __restrict__ ie,
                        float4* __restrict__ x) {
    int i = blockIdx.x * blockDim.x + threadIdx.x;          // over N*EMB/4
    if (i >= N_NODES * EMB / 4) return;
    const int UQ = U_NODES * EMB / 4;
    x[i] = (i < UQ) ? ue[i] : ie[i - UQ];
}

// smoo = x - gamma*(x - x0); also zero the scatter accumulator
__global__ void smoo_zero4(const float4* __restrict__ x,
                           const float4* __restrict__ ue, const float4* __restrict__ ie,
                           float4* __restrict__ smoo, float4* __restrict__ ctz) {
    int i = blockIdx.x * blockDim.x + threadIdx.x;          // over N*EMB/4
    if (i >= N_NODES * EMB / 4) return;
    const int UQ = U_NODES * EMB / 4;
    float4 x0 = (i < UQ) ? ue[i] : ie[i - UQ];
    float4 xv = x[i];
    float4 s;
    s.x = xv.x - GAMMA * (xv.x - x0.x);
    s.y = xv.y - GAMMA * (xv.y - x0.y);
    s.z = xv.z - GAMMA * (xv.z - x0.z);
    s.w = xv.w - GAMMA * (xv.w - x0.w);
    smoo[i] = s;
    ctz[i]  = make_float4(0.f, 0.f, 0.f, 0.f);
}

__device__ __forceinline__ float clip_lam(float v) {
    return fminf(LAM, fmaxf(-LAM, v));
}

// one thread per (edge, 4 dims): b128 gather of smoo, b128 z load/store,
// dual update+clip, 8 scalar f32 scatter-atomics (L2-resident accumulator)
__global__ void edge_step4(const float* __restrict__ smoo, const float* __restrict__ dinv,
                           const int* __restrict__ eu, const int* __restrict__ ei,
                           float4* __restrict__ z, float* __restrict__ ctz, int first) {
    int gid = blockIdx.x * blockDim.x + threadIdx.x;        // over NEDGES * 16
    if (gid >= NEDGES * (EMB / 4)) return;
    int e  = gid >> 4;
    int dq = (gid & 15) << 2;                               // dim quad base: 0,4,...,60
    int src = U_NODES + ei[e];                              // +1 endpoint (item)
    int dst = eu[e];                                        // -1 endpoint (user)
    float ss = dinv[src];
    float sd = dinv[dst];

    float4 sv = *(const float4*)(smoo + src * EMB + dq);    // global_load_b128
    float4 dv = *(const float4*)(smoo + dst * EMB + dq);    // global_load_b128
    float4 zo = first ? make_float4(0.f, 0.f, 0.f, 0.f) : z[gid];

    float4 zn;
    zn.x = clip_lam(zo.x + COEF * (ss * sv.x - sd * dv.x));
    zn.y = clip_lam(zo.y + COEF * (ss * sv.y - sd * dv.y));
    zn.z = clip_lam(zo.z + COEF * (ss * sv.z - sd * dv.z));
    zn.w = clip_lam(zo.w + COEF * (ss * sv.w - sd * dv.w));
    z[gid] = zn;                                            // global_store_b128

    float* cs = ctz + src * EMB + dq;
    float* cd = ctz + dst * EMB + dq;
    atomicAdd(cs + 0,  ss * zn.x);
    atomicAdd(cs + 1,  ss * zn.y);
    atomicAdd(cs + 2,  ss * zn.z);
    atomicAdd(cs + 3,  ss * zn.w);
    atomicAdd(cd + 0, -sd * zn.x);
    atomicAdd(cd + 1, -sd * zn.y);
    atomicAdd(cd + 2, -sd * zn.z);
    atomicAdd(cd + 3, -sd * zn.w);
}

__global__ void combine4(const float4* __restrict__ smoo, const float4* __restrict__ ctz,
                         float4* __restrict__ x) {
    int i = blockIdx.x * blockDim.x + threadIdx.x;          // over N*EMB/4
    if (i >= N_NODES * EMB / 4) return;
    float4 s = smoo[i];
    float4 c = ctz[i];
    float4 r;
    r.x = s.x - GAMMA * c.x;
    r.y = s.y - GAMMA * c.y;
    r.z = s.z - GAMMA * c.z;
    r.w = s.w - GAMMA * c.w;
    x[i] = r;
}

// ---- final batched dot via V_WMMA_F32_16X16X4_F32 ----------------------------
// One wave computes 16 dot products: D = A(16x64 users) x B(64x16 items),
// accumulated over 16 K=4 steps; the diagonal of the 16x16 D tile is the
// per-batch-element dot. A layout (ISA 7.12.2, 32-bit A 16x4): lane L holds
// row M = L%16; VGPR v holds K = v + 2*(L/16). B is N-symmetric, so both
// fragments index row (L%16). D layout: VGPR r, lane L -> (M = r+8*(L/16),
// N = L%16); diagonal lives at lanes 0..7 (VGPR=lane) and lanes 24..31 (VGPR=lane-24).
__global__ void dot_wmma(const float* __restrict__ x,
                         const int* __restrict__ users, const int* __restrict__ items,
                         float* __restrict__ out) {
    int lane = threadIdx.x & 31;
    int wave = threadIdx.x >> 5;
    int base = (blockIdx.x * (blockDim.x >> 5) + wave) * 16;   // uniform per wave

    int m  = lane & 15;
    int u  = users[base + m];
    int it = items[base + m];
    const float* urow = x + (long long)u * EMB;
    const float* irow = x + (long long)(U_NODES + it) * EMB;
    int khalf = (lane >> 4) << 1;                 // 0 for lanes 0-15, 2 for 16-31

    v8f c = {0.f, 0.f, 0.f, 0.f, 0.f, 0.f, 0.f, 0.f};
#pragma unroll
    for (int kb = 0; kb < EMB; kb += 4) {
        v2f a, b;
        a.x = urow[kb + khalf];
        a.y = urow[kb + khalf + 1];
        b.x = irow[kb + khalf];
        b.y = irow[kb + khalf + 1];
        // 8 args: (neg_a, A, neg_b, B, c_mod, C, reuse_a, reuse_b)
        c = __builtin_amdgcn_wmma_f32_16x16x4_f32(
                false, a, false, b, (short)0, c, false, false);
    }

    // extract diagonal of the 16x16 f32 D tile
    float v = 0.0f;
    int r = (lane < 16) ? lane : (lane - 24);
#pragma unroll
    for (int j = 0; j < 8; ++j)
        if (r == j) v = c[j];
    if (lane < 8)        out[base + lane] = v;            // M = N = lane
    else if (lane >= 24) out[base + 8 + (lane - 24)] = v; // M = N = 8+(lane-24)
}

// -----------------------------------------------------------------------------
extern "C" void kernel_launch(void* const* d_in, const int* in_sizes, int n_in,
                              void* d_out, int out_size, void* d_ws, size_t ws_size,
                              hipStream_t stream) {
    const float* ue    = (const float*)d_in[0];
    const float* ie    = (const float*)d_in[1];
    const int*   eu    = (const int*)d_in[2];
    const int*   ei    = (const int*)d_in[3];
    const int*   users = (const int*)d_in[4];
    const int*   items = (const int*)d_in[5];
    float*       out   = (float*)d_out;

    // workspace layout (floats): dinv | x | smoo | ctz | z  (all 16B aligned)
    float* ws = (float*)d_ws;
    size_t off = 0;
    float* dinv = ws + off; off += ((size_t)N_NODES + 255) / 256 * 256;
    float* x    = ws + off; off += (size_t)N_NODES * EMB;
    float* smoo = ws + off; off += (size_t)N_NODES * EMB;
    float* ctz  = ws + off; off += (size_t)N_NODES * EMB;
    float* z    = ws + off; off += (size_t)NEDGES * EMB;   // 256 MB dual variable

    const int B  = 256;
    const long long NQ = (long long)N_NODES * EMB / 4;     // node float4 count
    const long long EQ = (long long)NEDGES * (EMB / 4);    // edge quad count

    init_deg   <<<cdiv_h(N_NODES, B), B, 0, stream>>>(dinv);
    accum_deg  <<<cdiv_h(NEDGES, B),  B, 0, stream>>>(eu, ei, dinv);
    finish_dinv<<<cdiv_h(N_NODES, B), B, 0, stream>>>(dinv);
    init_x4    <<<cdiv_h(NQ, B),      B, 0, stream>>>(
        (const float4*)ue, (const float4*)ie, (float4*)x);

    for (int k = 0; k < KLAYERS; ++k) {
        smoo_zero4<<<cdiv_h(NQ, B), B, 0, stream>>>(
            (const float4*)x, (const float4*)ue, (const float4*)ie,
            (float4*)smoo, (float4*)ctz);
        edge_step4<<<cdiv_h(EQ, B), B, 0, stream>>>(
            smoo, dinv, eu, ei, (float4*)z, ctz, (k == 0) ? 1 : 0);
        combine4  <<<cdiv_h(NQ, B), B, 0, stream>>>(
            (const float4*)smoo, (const float4*)ctz, (float4*)x);
    }

    // 16 dots per wave, 8 waves per block -> 128 per block, 64 blocks = 8192
    dot_wmma<<<BATCHSZ / 128, 256, 0, stream>>>(x, users, items, out);
}